// GATModel_62036507623982
// MI455X (gfx1250) — compile-verified
//
#include <hip/hip_runtime.h>

// ---------------------------------------------------------------------------
// 3-layer GAT (PyG GATConv semantics) for MI455X (gfx1250).
//   L1: [N,128] x W1[128,128] -> H=2, C=64,  D=128
//   L2: [N,128] x W2[128,256] -> H=2, C=128, D=256
//   L3: [N,256] x W3[256,256] -> H=2, C=128, D=256
// GEMMs use V_WMMA_F32_16X16X4_F32 (native fp32 WMMA, exact precision).
// Edge softmax/aggregation is HBM/L2-atomic bound; scatter targets (<=51MB)
// are L2-resident on the 192MB L2.
// ---------------------------------------------------------------------------

#define N_NODES 50000
#define N_EDGES 800000
#define ET      (N_EDGES + N_NODES)   // self-loops appended
#define NEG_SLOPE 0.2f

typedef __attribute__((ext_vector_type(2))) float v2f;
typedef __attribute__((ext_vector_type(8))) float v8f;

// Monotonic float <-> uint encoding for atomicMax on floats (handles negatives).
__device__ __forceinline__ unsigned fenc(float f) {
  unsigned u = __float_as_uint(f);
  return (u & 0x80000000u) ? ~u : (u | 0x80000000u);
}
__device__ __forceinline__ float fdec(unsigned u) {
  return __uint_as_float((u & 0x80000000u) ? (u & 0x7FFFFFFFu) : ~u);
}
#define ENC_NEG_INF 0x007FFFFFu   // fenc(-inf)

// ------------------------- dense GEMM: H = X @ W ---------------------------
// One wave computes one 16x16 tile of H. M is a multiple of 16 (50000=3125*16).
// A frag (16x4 f32): lane l holds X[m=l&15][k + 2*(l>>4)] , [.. + 1]
// B frag (4x16 f32): lane l holds W[k + 2*(l>>4)][n=l&15] , W[k+2*(l>>4)+1][n]
// D (16x16 f32):     VGPR i -> row M = i + 8*(l>>4), col N = l&15
__global__ void gemm_wmma_f32(const float* __restrict__ X,
                              const float* __restrict__ W,
                              float* __restrict__ Hout,
                              int M, int K, int Nd) {
  const int lane = threadIdx.x & 31;
  const int wid  = (int)((blockIdx.x * blockDim.x + threadIdx.x) >> 5);
  const int ntile = Nd >> 4;
  if (wid >= (M >> 4) * ntile) return;
  const int mt = wid / ntile;
  const int nt = wid - mt * ntile;
  const int lrow = lane & 15;
  const int lhi  = lane >> 4;

  const float* ap = X + (size_t)(mt * 16 + lrow) * K + 2 * lhi;
  const float* bp = W + (size_t)(2 * lhi) * Nd + nt * 16 + lrow;

  v8f acc = {};
  for (int k = 0; k < K; k += 4) {
    v2f a, b;
    a.x = ap[k];
    a.y = ap[k + 1];
    b.x = bp[(size_t)k * Nd];
    b.y = bp[(size_t)(k + 1) * Nd];
    acc = __builtin_amdgcn_wmma_f32_16x16x4_f32(false, a, false, b,
                                                (short)0, acc, false, false);
  }
  float* op = Hout + (size_t)(mt * 16 + 8 * lhi) * Nd + nt * 16 + lrow;
#pragma unroll
  for (int i = 0; i < 8; ++i) op[(size_t)i * Nd] = acc[i];
}

// --------------- per-node attention logits: alpha_s / alpha_d --------------
// alpha_s[i][h] = sum_c H[i][h*C+c] * a_src[h*C+c]   (wave per node)
__global__ void node_alpha(const float* __restrict__ Hf,
                           const float* __restrict__ asrc,
                           const float* __restrict__ adst,
                           float* __restrict__ als, float* __restrict__ ald,
                           int C) {
  const int lane = threadIdx.x & 31;
  const int node = (int)((blockIdx.x * blockDim.x + threadIdx.x) >> 5);
  if (node >= N_NODES) return;
  const float* hp = Hf + (size_t)node * (2 * C);
#pragma unroll
  for (int h = 0; h < 2; ++h) {
    float s = 0.f, d = 0.f;
    for (int c = lane; c < C; c += 32) {
      float v = hp[h * C + c];
      s += v * asrc[h * C + c];
      d += v * adst[h * C + c];
    }
    for (int off = 16; off > 0; off >>= 1) {
      s += __shfl_down(s, off);
      d += __shfl_down(d, off);
    }
    if (lane == 0) { als[node * 2 + h] = s; ald[node * 2 + h] = d; }
  }
}

__global__ void init_softmax(unsigned* __restrict__ mx, float* __restrict__ dn,
                             int n) {
  int i = blockIdx.x * blockDim.x + threadIdx.x;
  if (i < n) { mx[i] = ENC_NEG_INF; dn[i] = 0.f; }
}

__device__ __forceinline__ void edge_sd(const long long* ei, int e, int& s, int& d) {
  if (e < N_EDGES) { s = (int)ei[e]; d = (int)ei[N_EDGES + e]; }
  else             { s = d = e - N_EDGES; }   // appended self-loop
}

// e = leaky_relu(as[src] + ad[dst]); segment max over dst (atomicMax, encoded)
__global__ void edge_max(const long long* __restrict__ ei,
                         const float* __restrict__ als,
                         const float* __restrict__ ald,
                         float* __restrict__ ebuf, unsigned* __restrict__ mx) {
  int e = blockIdx.x * blockDim.x + threadIdx.x;
  if (e >= ET) return;
  int s, d; edge_sd(ei, e, s, d);
#pragma unroll
  for (int h = 0; h < 2; ++h) {
    float v = als[s * 2 + h] + ald[d * 2 + h];
    v = (v > 0.f) ? v : NEG_SLOPE * v;
    ebuf[(size_t)e * 2 + h] = v;
    atomicMax(&mx[d * 2 + h], fenc(v));
  }
}

// ex = exp(e - max[dst]); segment sum over dst
__global__ void edge_expsum(const long long* __restrict__ ei,
                            float* __restrict__ ebuf,
                            const unsigned* __restrict__ mx,
                            float* __restrict__ dn) {
  int e = blockIdx.x * blockDim.x + threadIdx.x;
  if (e >= ET) return;
  int s, d; edge_sd(ei, e, s, d);
  (void)s;
#pragma unroll
  for (int h = 0; h < 2; ++h) {
    float ex = expf(ebuf[(size_t)e * 2 + h] - fdec(mx[d * 2 + h]));
    ebuf[(size_t)e * 2 + h] = ex;
    atomicAdd(&dn[d * 2 + h], ex);
  }
}

// out[i][:] = bias  (D is a power of two: 128 or 256)
__global__ void init_bias(float* __restrict__ Out, const float* __restrict__ b,
                          int D) {
  int i = blockIdx.x * blockDim.x + threadIdx.x;
  if (i < N_NODES * D) Out[i] = b[i & (D - 1)];
}

// alpha = ex / denom[dst]; out[dst] += h[src] * alpha   (wave per edge)
__global__ void edge_aggregate(const long long* __restrict__ ei,
                               const float* __restrict__ ebuf,
                               const float* __restrict__ dn,
                               const float* __restrict__ Hf,
                               float* __restrict__ Out,
                               float* __restrict__ alpha_out, int C) {
  const int lane = threadIdx.x & 31;
  const int e = (int)((blockIdx.x * blockDim.x + threadIdx.x) >> 5);
  if (e >= ET) return;
  int s, d; edge_sd(ei, e, s, d);
  const float a0 = ebuf[(size_t)e * 2 + 0] / dn[d * 2 + 0];
  const float a1 = ebuf[(size_t)e * 2 + 1] / dn[d * 2 + 1];
  if (lane == 0) {
    alpha_out[(size_t)e * 2 + 0] = a0;
    alpha_out[(size_t)e * 2 + 1] = a1;
  }
  const int D = 2 * C;
  const float* hp = Hf + (size_t)s * D;
  float* op = Out + (size_t)d * D;
  for (int c = lane; c < D; c += 32)
    atomicAdd(&op[c], hp[c] * ((c < C) ? a0 : a1));
}

__global__ void relu_k(float* __restrict__ x, int n) {
  int i = blockIdx.x * blockDim.x + threadIdx.x;
  if (i < n) x[i] = fmaxf(x[i], 0.f);
}

// edge_index with self-loops, written (as float) into all 3 output slots
__global__ void write_ei(const long long* __restrict__ ei, float* __restrict__ o1,
                         float* __restrict__ o2, float* __restrict__ o3) {
  int e = blockIdx.x * blockDim.x + threadIdx.x;
  if (e >= ET) return;
  int s, d; edge_sd(ei, e, s, d);
  float fs = (float)s, fd = (float)d;
  o1[e] = fs; o1[ET + e] = fd;
  o2[e] = fs; o2[ET + e] = fd;
  o3[e] = fs; o3[ET + e] = fd;
}

// ---------------------------------------------------------------------------
extern "C" void kernel_launch(void* const* d_in, const int* in_sizes, int n_in,
                              void* d_out, int out_size, void* d_ws,
                              size_t ws_size, hipStream_t stream) {
  (void)in_sizes; (void)n_in; (void)out_size; (void)ws_size;

  const float*     x   = (const float*)d_in[0];
  const long long* ei  = (const long long*)d_in[1];
  const float* W1  = (const float*)d_in[2];
  const float* as1 = (const float*)d_in[3];
  const float* ad1 = (const float*)d_in[4];
  const float* b1  = (const float*)d_in[5];
  const float* W2  = (const float*)d_in[6];
  const float* as2 = (const float*)d_in[7];
  const float* ad2 = (const float*)d_in[8];
  const float* b2  = (const float*)d_in[9];
  const float* W3  = (const float*)d_in[10];
  const float* as3 = (const float*)d_in[11];
  const float* ad3 = (const float*)d_in[12];
  const float* b3  = (const float*)d_in[13];

  // ---- output layout: h3, (ei,a1), (ei,a2), (ei,a3), all flat float ----
  float* out   = (float*)d_out;
  float* out_h = out;
  float* ei1 = out_h + (size_t)N_NODES * 256;
  float* a1o = ei1 + 2 * (size_t)ET;
  float* ei2 = a1o + 2 * (size_t)ET;
  float* a2o = ei2 + 2 * (size_t)ET;
  float* ei3 = a2o + 2 * (size_t)ET;
  float* a3o = ei3 + 2 * (size_t)ET;

  // ---- workspace layout ----
  float* Hf  = (float*)d_ws;                     // [N,256] post-GEMM features
  float* Of  = Hf + (size_t)N_NODES * 256;       // [N,256] aggregated output
  float* als = Of + (size_t)N_NODES * 256;       // [N,2]
  float* ald = als + N_NODES * 2;                // [N,2]
  float* dn  = ald + N_NODES * 2;                // [N,2]
  unsigned* mx = (unsigned*)(dn + N_NODES * 2);  // [N,2]
  float* eb  = (float*)(mx + N_NODES * 2);       // [ET,2]

  write_ei<<<(ET + 255) / 256, 256, 0, stream>>>(ei, ei1, ei2, ei3);

  auto layer = [&](const float* Xin, int K, int C, const float* W,
                   const float* asc, const float* adc, const float* bias,
                   float* Outp, float* alphao, bool do_relu) {
    const int D = 2 * C;
    const int tiles = (N_NODES / 16) * (D / 16);
    gemm_wmma_f32<<<(tiles + 7) / 8, 256, 0, stream>>>(Xin, W, Hf, N_NODES, K, D);
    node_alpha<<<(N_NODES * 32 + 255) / 256, 256, 0, stream>>>(Hf, asc, adc, als, ald, C);
    init_softmax<<<(N_NODES * 2 + 255) / 256, 256, 0, stream>>>(mx, dn, N_NODES * 2);
    edge_max<<<(ET + 255) / 256, 256, 0, stream>>>(ei, als, ald, eb, mx);
    edge_expsum<<<(ET + 255) / 256, 256, 0, stream>>>(ei, eb, mx, dn);
    init_bias<<<(N_NODES * D + 255) / 256, 256, 0, stream>>>(Outp, bias, D);
    edge_aggregate<<<((size_t)ET * 32 + 255) / 256, 256, 0, stream>>>(ei, eb, dn, Hf, Outp, alphao, C);
    if (do_relu)
      relu_k<<<(N_NODES * D + 255) / 256, 256, 0, stream>>>(Outp, N_NODES * D);
  };

  // Of is only read by the first kernel (GEMM) of each layer, so it can be
  // safely reused as that layer's aggregation target (stream-ordered).
  layer(x,  128,  64, W1, as1, ad1, b1, Of,    a1o, true);
  layer(Of, 128, 128, W2, as2, ad2, b2, Of,    a2o, true);
  layer(Of, 256, 128, W3, as3, ad3, b3, out_h, a3o, false);
}